// SegGmm_75660143886580
// MI455X (gfx1250) — compile-verified
//
#include <hip/hip_runtime.h>
#include <math.h>

typedef float v2f __attribute__((ext_vector_type(2)));
typedef float v8f __attribute__((ext_vector_type(8)));

#define EPS 1e-8f
#define LOG2E 1.4426950408889634f

__device__ __forceinline__ float fast_rcp(float v)  { return __builtin_amdgcn_rcpf(v); }
__device__ __forceinline__ float fast_sqrt(float v) { return __builtin_amdgcn_sqrtf(v); }
__device__ __forceinline__ float fast_exp(float v)  { return __builtin_amdgcn_exp2f(v * LOG2E); }

__device__ __forceinline__ float fast_tanh(float v) {
#if __has_builtin(__builtin_amdgcn_tanhf)
    return __builtin_amdgcn_tanhf(v);          // native V_TANH_F32 (CDNA5 trans op)
#else
    // branch-free: tanh(v) = 1 - 2/(exp2(2*log2(e)*v) + 1)
    float e = __builtin_amdgcn_exp2f(v * (2.0f * LOG2E));
    return 1.0f - 2.0f * __builtin_amdgcn_rcpf(e + 1.0f);
#endif
}

// One thread = one point for all element-wise work. Per wave (32 lanes), the
// estimation-net first layer (z[16x3] @ W_est1^T[3x10] + b) runs as two
// V_WMMA_F32_16X16X4_F32 tiles (points 0-15 and 16-31 of the wave), K padded
// 3->4, N padded 10->16, bias folded into the C accumulator.
__global__ __launch_bounds__(256) void dagmm_fwd_kernel(
    const float* __restrict__ x,
    const float* __restrict__ W_enc,  const float* __restrict__ b_enc,
    const float* __restrict__ W_dec,  const float* __restrict__ b_dec,
    const float* __restrict__ W_est1, const float* __restrict__ b_est1,
    const float* __restrict__ W_est2, const float* __restrict__ b_est2,
    float* __restrict__ out, int N)
{
    // per-wave 32x17 padded transpose tile (stride 17 -> bank-conflict free)
    __shared__ float lds[8 * 32 * 17];

    const int tid  = threadIdx.x;
    const int lane = tid & 31;
    const int wave = tid >> 5;
    const int i    = blockIdx.x * 256 + tid;
    const int iL   = (i < N) ? i : (N - 1);   // clamp loads, keep EXEC full

    // ---------------- element-wise encoder / decoder / recon features -------
    const float2 xv = ((const float2*)x)[iL];         // global_load_b64
    const float x0 = xv.x, x1 = xv.y;

    const float we0 = W_enc[0], we1 = W_enc[1], be  = b_enc[0];
    const float wd0 = W_dec[0], wd1 = W_dec[1];
    const float bd0 = b_dec[0], bd1 = b_dec[1];

    const float enc  = fmaf(x1, we1, fmaf(x0, we0, be));
    const float dec0 = fmaf(enc, wd0, bd0);
    const float dec1 = fmaf(enc, wd1, bd1);

    const float xn   = fast_sqrt(x0 * x0 + x1 * x1);
    const float dn   = fast_sqrt(dec0 * dec0 + dec1 * dec1);
    const float dotv = x0 * dec0 + x1 * dec1;
    const float rec_cos = dotv * fast_rcp(fmaxf(xn, EPS) * fmaxf(dn, EPS));
    const float ex = x0 - dec0, ey = x1 - dec1;
    const float rec_euc = fast_sqrt(ex * ex + ey * ey) * fast_rcp(xn);

    const float z0 = enc, z1 = rec_euc, z2 = rec_cos;   // z = [enc, euc, cos]

    // ---------------- layer 1 (3 -> 10) via V_WMMA_F32_16X16X4_F32 ---------
    const int  ncol = lane & 15;
    const bool hi   = (lane >= 16);

    // B[k][n] = W_est1[n][k]; 4x16 layout: v0 = rows K=0 (lanes 0-15) / K=2
    // (lanes 16-31), v1 = rows K=1 / K=3 (K=3 is zero pad), n>=10 zero pad.
    float bv0 = 0.0f, bv1 = 0.0f;
    if (ncol < 10) {
        bv0 = hi ? W_est1[ncol * 3 + 2] : W_est1[ncol * 3 + 0];
        bv1 = hi ? 0.0f                 : W_est1[ncol * 3 + 1];
    }
    v2f bb; bb.x = bv0; bb.y = bv1;

    // C = bias broadcast down each column n
    const float cb = (ncol < 10) ? b_est1[ncol] : 0.0f;
    v8f c0 = {cb, cb, cb, cb, cb, cb, cb, cb};

    // A tiles, 16x4 layout: lanes 0-15 hold K=0 (v0), K=1 (v1) for row M=lane;
    // lanes 16-31 hold K=2 (v0), K=3=pad (v1) for row M=lane-16.
    const float z2lo = __shfl(z2, ncol,      32);  // z2 of point (lane&15)
    const float z0hi = __shfl(z0, ncol + 16, 32);  // z0 of point (lane&15)+16
    const float z1hi = __shfl(z1, ncol + 16, 32);

    v2f a_t0, a_t1;
    a_t0.x = hi ? z2lo : z0;   a_t0.y = hi ? 0.0f : z1;    // points 0..15
    a_t1.x = hi ? z2   : z0hi; a_t1.y = hi ? 0.0f : z1hi;  // points 16..31

    v8f d_t0 = __builtin_amdgcn_wmma_f32_16x16x4_f32(
        false, a_t0, false, bb, (short)0, c0, false, false);
    v8f d_t1 = __builtin_amdgcn_wmma_f32_16x16x4_f32(
        false, a_t1, false, bb, (short)0, c0, false, false);

    // ---------------- tanh + LDS transpose back to per-point layout --------
    // D layout: VGPR v holds rows M=v (lanes 0-15) and M=v+8 (lanes 16-31),
    // column N = lane&15.
    float* hb = &lds[wave * (32 * 17)];
    const int mb = hi ? 8 : 0;
#pragma unroll
    for (int v = 0; v < 8; ++v) {
        hb[(mb + v) * 17 + ncol]      = fast_tanh(d_t0[v]);
        hb[(16 + mb + v) * 17 + ncol] = fast_tanh(d_t1[v]);
    }
    __syncthreads();

    float h[10];
#pragma unroll
    for (int j = 0; j < 10; ++j) h[j] = hb[lane * 17 + j];

    // ---------------- layer 2 (10 -> 2) + softmax, per lane ----------------
    float e0 = b_est2[0], e1 = b_est2[1];
#pragma unroll
    for (int j = 0; j < 10; ++j) {
        e0 = fmaf(W_est2[j],      h[j], e0);
        e1 = fmaf(W_est2[10 + j], h[j], e1);
    }
    const float mx  = fmaxf(e0, e1);
    const float g0  = fast_exp(e0 - mx);
    const float g1  = fast_exp(e1 - mx);
    const float inv = fast_rcp(g0 + g1);

    // ---------------- stores (predicated; outputs concatenated flat) -------
    if (i < N) {
        out[i] = enc;                                    // enc  [N,1] @ 0
        float2 dv; dv.x = dec0; dv.y = dec1;             // dec  [N,2] @ N
        ((float2*)(out + N))[i] = dv;
        float* zp = out + 3 * N + 3 * i;                 // z    [N,3] @ 3N
        zp[0] = z0; zp[1] = z1; zp[2] = z2;
        float2 gv; gv.x = g0 * inv; gv.y = g1 * inv;     // gamma[N,2] @ 6N
        ((float2*)(out + 6 * N))[i] = gv;
    }
}

extern "C" void kernel_launch(void* const* d_in, const int* in_sizes, int n_in,
                              void* d_out, int out_size, void* d_ws, size_t ws_size,
                              hipStream_t stream) {
    const float* x      = (const float*)d_in[0];
    const float* W_enc  = (const float*)d_in[1];
    const float* b_enc  = (const float*)d_in[2];
    const float* W_dec  = (const float*)d_in[3];
    const float* b_dec  = (const float*)d_in[4];
    const float* W_est1 = (const float*)d_in[5];
    const float* b_est1 = (const float*)d_in[6];
    const float* W_est2 = (const float*)d_in[7];
    const float* b_est2 = (const float*)d_in[8];

    const int N = in_sizes[0] / 2;
    const int blocks = (N + 255) / 256;

    dagmm_fwd_kernel<<<blocks, 256, 0, stream>>>(
        x, W_enc, b_enc, W_dec, b_dec, W_est1, b_est1, W_est2, b_est2,
        (float*)d_out, N);
}